// HypergraphConv_49185965473981
// MI455X (gfx1250) — compile-verified
//
#include <hip/hip_runtime.h>
#include <math.h>

typedef __attribute__((ext_vector_type(2))) float v2f;
typedef __attribute__((ext_vector_type(8))) float v8f;

#define DIN   128
#define DOUTC 128
#define KEDGE 16
#define EPSF  1e-8f
#define WREC  0.0625f                 /* 1/16 */
#define DEINV (1.0f / (1.0f + 1e-8f)) /* 1/(De+eps), De == 1 exactly */

// ---------------------------------------------------------------- zero fill
__global__ void zero_f32_kernel(float* __restrict__ p, size_t n) {
    size_t i = (size_t)blockIdx.x * blockDim.x + threadIdx.x;
    size_t stride = (size_t)gridDim.x * blockDim.x;
    for (; i < n; i += stride) p[i] = 0.0f;
}

// ---------------------------------------------------- node degree histogram
__global__ void count_kernel(const int* __restrict__ en, int* __restrict__ cnt,
                             int total) {
    int i = blockIdx.x * blockDim.x + threadIdx.x;
    if (i < total) atomicAdd(&cnt[en[i]], 1);
}

// ------------------------------------------------- out = x @ W via WMMA f32
// One block = 16 node rows. 8 waves, each wave owns a 16-wide column panel.
// B (W panel, 128x16) lives in VGPRs; x tile in padded LDS.
__global__ __launch_bounds__(256) void gemm_wmma_kernel(
    const float* __restrict__ x, const float* __restrict__ W,
    float* __restrict__ out, int N) {
    __shared__ float sX[16 * 129];  // +1 pad: avoid 16-way LDS bank conflict
    const int tid  = threadIdx.x;
    const int row0 = blockIdx.x * 16;

    for (int i = tid; i < 16 * DIN; i += 256) {
        int r = i >> 7, c0 = i & 127;
        int rr = row0 + r; if (rr >= N) rr = N - 1;
        sX[r * 129 + c0] = x[(size_t)rr * DIN + c0];
    }
    __syncthreads();

    const int lane    = tid & 31;
    const int ln      = lane & 15;          // N / M index within half
    const int kb      = (lane >> 4) << 1;   // K pair select (0 or 2)
    const int colbase = (tid >> 5) * 16;    // wave -> column panel

    // Preload B panel (128 x 16 of W) into registers: 32 x v2f = 64 VGPRs
    v2f breg[32];
#pragma unroll
    for (int i = 0; i < 32; ++i) {
        int k0 = i * 4;
        breg[i].x = W[(size_t)(k0 + kb) * DOUTC + colbase + ln];
        breg[i].y = W[(size_t)(k0 + kb + 1) * DOUTC + colbase + ln];
    }

    v8f c = {};
#pragma unroll
    for (int i = 0; i < 32; ++i) {
        int k0 = i * 4;
        v2f a;
        a.x = sX[ln * 129 + k0 + kb];
        a.y = sX[ln * 129 + k0 + kb + 1];
        // D = A(16x4) * B(4x16) + C  -- v_wmma_f32_16x16x4_f32
        c = __builtin_amdgcn_wmma_f32_16x16x4_f32(
            false, a, false, breg[i], (short)0, c, false, false);
    }

    const int rowHalf = (lane >> 4) << 3;  // lanes 16-31 hold M=8..15
#pragma unroll
    for (int j = 0; j < 8; ++j) {
        int rr = row0 + rowHalf + j;
        if (rr < N) out[(size_t)rr * DOUTC + colbase + ln] = c[j];
    }
}

// -------------------------------------- deterministic column-sum of x (mean)
__global__ void colsum_kernel(const float* __restrict__ x,
                              float* __restrict__ colpart, int N) {
    int d  = threadIdx.x;  // 128 threads = 128 columns
    int r0 = blockIdx.x * 256;
    int r1 = r0 + 256; if (r1 > N) r1 = N;
    float s = 0.f;
    for (int r = r0; r < r1; ++r) s += x[(size_t)r * DIN + d];
    colpart[(size_t)blockIdx.x * DIN + d] = s;
}

__global__ void alpha_kernel(const float* __restrict__ colpart, int CB,
                             const float* __restrict__ gw,
                             const float* __restrict__ gb,
                             float* __restrict__ alphap, int N) {
    __shared__ float red[128];
    int d = threadIdx.x;
    float s = 0.f;
    for (int b = 0; b < CB; ++b) s += colpart[(size_t)b * DIN + d];
    red[d] = (s / (float)N) * gw[d];
    __syncthreads();
    for (int off = 64; off > 0; off >>= 1) {
        if (d < off) red[d] += red[d + off];
        __syncthreads();
    }
    if (d == 0) {
        float v = red[0] + gb[0];
        alphap[0] = 1.0f / (1.0f + __expf(-v));
    }
}

// ------------------------- fused per-edge gather (H^T) + scatter-add (H)
__global__ __launch_bounds__(128) void edge_kernel(
    const int* __restrict__ en, const float* __restrict__ outf,
    const int* __restrict__ cnt, float* __restrict__ hyper) {
    __shared__ int   snode[KEDGE];
    __shared__ float sdvis[KEDGE];
    const int e = blockIdx.x;
    const int t = threadIdx.x;  // column 0..127
    if (t < KEDGE) {
        int nd   = en[(size_t)e * KEDGE + t];
        snode[t] = nd;
        sdvis[t] = rsqrtf((float)cnt[nd] * WREC + EPSF);  // Dv^-1/2
    }
    __syncthreads();
    float acc = 0.f;
#pragma unroll
    for (int k = 0; k < KEDGE; ++k)
        acc += outf[(size_t)snode[k] * DOUTC + t] * sdvis[k];
    float y = acc * WREC * DEINV;  // w * H^T(scaled), then De^-1
#pragma unroll
    for (int k = 0; k < KEDGE; ++k)
        atomicAdd(&hyper[(size_t)snode[k] * DOUTC + t], y);
}

// ----------------------------- final blend: alpha*out + (1-a)*hyper + bias
__global__ void finalize_kernel(const float* __restrict__ out_ws,
                                const int* __restrict__ cnt,
                                const float* __restrict__ bias,
                                const float* __restrict__ alphap,
                                float* __restrict__ y, size_t total) {
    size_t i = (size_t)blockIdx.x * blockDim.x + threadIdx.x;
    if (i >= total) return;
    size_t n = i >> 7;
    int    d = (int)(i & 127);
    float dvis      = rsqrtf((float)cnt[n] * WREC + EPSF);
    float alpha     = alphap[0];
    float hyper_out = WREC * dvis * y[i];  // w * Dv^-1/2 * accumulated
    y[i] = alpha * out_ws[i] + (1.0f - alpha) * hyper_out + bias[d];
}

extern "C" void kernel_launch(void* const* d_in, const int* in_sizes, int n_in,
                              void* d_out, int out_size, void* d_ws,
                              size_t ws_size, hipStream_t stream) {
    (void)n_in; (void)out_size; (void)ws_size;
    const float* x    = (const float*)d_in[0];
    const int*   en   = (const int*)d_in[1];
    const float* W    = (const float*)d_in[2];
    const float* bias = (const float*)d_in[3];
    const float* gw   = (const float*)d_in[4];
    const float* gb   = (const float*)d_in[5];

    const int N  = in_sizes[0] / DIN;    // 200000
    const int E  = in_sizes[1] / KEDGE;  // 50000
    const int CB = (N + 255) / 256;      // colsum partial blocks

    // workspace layout: [out N*128 f32][cnt N i32][colpart CB*128 f32][alpha]
    float* out_ws  = (float*)d_ws;
    int*   cnt     = (int*)(out_ws + (size_t)N * DOUTC);
    float* colpart = (float*)(cnt + N);
    float* alphap  = colpart + (size_t)CB * DIN;
    float* hyper   = (float*)d_out;  // accumulate H(...) into d_out, finalize in place

    const size_t totalOut = (size_t)N * DOUTC;

    zero_f32_kernel<<<2048, 256, 0, stream>>>(hyper, totalOut);
    zero_f32_kernel<<<512, 256, 0, stream>>>(
        (float*)cnt, (size_t)N + (size_t)CB * DIN + 1);

    count_kernel<<<(E * KEDGE + 255) / 256, 256, 0, stream>>>(en, cnt, E * KEDGE);
    gemm_wmma_kernel<<<(N + 15) / 16, 256, 0, stream>>>(x, W, out_ws, N);
    colsum_kernel<<<CB, 128, 0, stream>>>(x, colpart, N);
    alpha_kernel<<<1, 128, 0, stream>>>(colpart, CB, gw, gb, alphap, N);
    edge_kernel<<<E, 128, 0, stream>>>(en, out_ws, cnt, hyper);
    finalize_kernel<<<(int)((totalOut + 255) / 256), 256, 0, stream>>>(
        out_ws, cnt, bias, alphap, hyper, totalOut);
}